// HungarianMSELoss_57312043597988
// MI455X (gfx1250) — compile-verified
//
#include <hip/hip_runtime.h>
#include <hip/hip_bf16.h>
#include <stdint.h>

// HungarianMSELoss == sort both rows, MSE of sorted pairs (rearrangement inequality).
// B=8192 rows, N=2048 elements/row. One workgroup per row.
// CDNA5 path: global_load_async_to_lds_b128 (ASYNCcnt) -> LDS bitonic sort
// (wave-synchronous inner stages, wave32 in-order DS) -> deterministic reduce.

#define ROW_N   2048
#define THREADS 256
#define WAVE    32
#define NWAVES  (THREADS / WAVE)   // 8
#define WIN     (ROW_N / NWAVES)   // 256-element wave-private window

// Compare-exchange pairs inside a wave-owned, WIN-aligned window. j <= WIN/2,
// so partner i|j stays in the window. Direction from the GLOBAL index i.
__device__ __forceinline__ void wave_ce(float* a, unsigned base, unsigned lane,
                                        unsigned k, unsigned j) {
    #pragma unroll
    for (unsigned t = lane; t < WIN / 2; t += WAVE) {
        unsigned io = ((t & ~(j - 1u)) << 1) | (t & (j - 1u)); // offset in window
        unsigned i  = base + io;
        unsigned p  = i | j;
        bool up = ((i & k) == 0u);
        float x = a[i], y = a[p];
        if ((x > y) == up) { a[i] = y; a[p] = x; }
    }
}

// All merge substeps j = jstart .. 1 inside the wave's window: no hardware
// barrier needed (same-wave LDS ops are in-order; one PC per wave32). Only a
// compiler scheduling fence between steps.
__device__ __forceinline__ void wave_substeps(float* sp, float* st, unsigned base,
                                              unsigned lane, unsigned k,
                                              unsigned jstart) {
    for (unsigned j = jstart; j > 0; j >>= 1) {
        wave_ce(sp, base, lane, k, j);
        wave_ce(st, base, lane, k, j);
        __builtin_amdgcn_wave_barrier();
        asm volatile("" ::: "memory");
    }
}

__global__ __launch_bounds__(THREADS)
void hungarian_sortmse_row_kernel(const float* __restrict__ pred,
                                  const float* __restrict__ targ,
                                  float* __restrict__ row_sums) {
    __shared__ float sp[ROW_N];
    __shared__ float st[ROW_N];

    const unsigned tid     = threadIdx.x;
    const unsigned lane    = tid & (WAVE - 1u);
    const unsigned wave_id = tid >> 5;
    const unsigned wbase   = wave_id * WIN;

    const float* gp = pred + (size_t)blockIdx.x * ROW_N;
    const float* gt = targ + (size_t)blockIdx.x * ROW_N;

    // ---- Async global -> LDS (B128 per lane; 512 segments / 256 lanes = 2 each/array)
    #pragma unroll
    for (int c = 0; c < 2; ++c) {
        unsigned elem  = (unsigned)(c * THREADS + tid) * 4u;   // float idx, 16B aligned
        unsigned lds_p = (unsigned)(uintptr_t)(&sp[elem]);     // low 32 bits = LDS offset
        unsigned lds_t = (unsigned)(uintptr_t)(&st[elem]);
        const float* ga = gp + elem;
        const float* gb = gt + elem;
        asm volatile("global_load_async_to_lds_b128 %0, %1, off"
                     :: "v"(lds_p), "v"(ga) : "memory");
        asm volatile("global_load_async_to_lds_b128 %0, %1, off"
                     :: "v"(lds_t), "v"(gb) : "memory");
    }
    asm volatile("s_wait_asynccnt 0" ::: "memory");
    __syncthreads();

    // ---- Phase 1: k = 2..WIN, fully wave-private (no block barriers).
    // After this, window w is sorted ascending/descending by parity of (wbase & k),
    // exactly the bitonic precondition for the cross-window merges.
    for (unsigned k = 2; k <= WIN; k <<= 1)
        wave_substeps(sp, st, wbase, lane, k, k >> 1);
    __syncthreads();

    // ---- Phase 2: k = 512..2048. Only j >= WIN crosses windows -> block barrier.
    for (unsigned k = 2 * WIN; k <= ROW_N; k <<= 1) {
        for (unsigned j = k >> 1; j >= WIN; j >>= 1) {
            #pragma unroll 4
            for (unsigned t = tid; t < ROW_N / 2; t += THREADS) {
                unsigned i = ((t & ~(j - 1u)) << 1) | (t & (j - 1u));
                unsigned p = i | j;
                bool up = ((i & k) == 0u);
                float a = sp[i], b = sp[p];
                if ((a > b) == up) { sp[i] = b; sp[p] = a; }
                float c0 = st[i], d0 = st[p];
                if ((c0 > d0) == up) { st[i] = d0; st[p] = c0; }
            }
            __syncthreads();
        }
        wave_substeps(sp, st, wbase, lane, k, WIN >> 1);
        __syncthreads();
    }

    // ---- Row sum of squared diffs ----
    float acc = 0.0f;
    #pragma unroll 4
    for (unsigned i = tid; i < ROW_N; i += THREADS) {
        float d = sp[i] - st[i];
        acc = fmaf(d, d, acc);
    }
    __syncthreads();
    sp[tid] = acc;
    __syncthreads();
    #pragma unroll
    for (unsigned s2 = THREADS / 2; s2 > 0; s2 >>= 1) {
        if (tid < s2) sp[tid] += sp[tid + s2];
        __syncthreads();
    }
    if (tid == 0) row_sums[blockIdx.x] = sp[0];
}

__global__ __launch_bounds__(THREADS)
void hungarian_final_reduce_kernel(const float* __restrict__ row_sums,
                                   float* __restrict__ out,
                                   int num_rows, float inv_total) {
    __shared__ float s[THREADS];
    float acc = 0.0f;
    for (int i = (int)threadIdx.x; i < num_rows; i += THREADS) acc += row_sums[i];
    s[threadIdx.x] = acc;
    __syncthreads();
    #pragma unroll
    for (unsigned s2 = THREADS / 2; s2 > 0; s2 >>= 1) {
        if (threadIdx.x < s2) s[threadIdx.x] += s[threadIdx.x + s2];
        __syncthreads();
    }
    if (threadIdx.x == 0) out[0] = s[0] * inv_total;
}

extern "C" void kernel_launch(void* const* d_in, const int* in_sizes, int n_in,
                              void* d_out, int out_size, void* d_ws, size_t ws_size,
                              hipStream_t stream) {
    const float* pred = (const float*)d_in[0];
    const float* targ = (const float*)d_in[1];
    float* out = (float*)d_out;

    const int total    = in_sizes[0];       // B * N
    const int num_rows = total / ROW_N;     // 8192

    float* row_sums = (float*)d_ws;         // num_rows floats of scratch

    hungarian_sortmse_row_kernel<<<num_rows, THREADS, 0, stream>>>(pred, targ, row_sums);

    const float inv_total = 1.0f / (float)((double)num_rows * (double)ROW_N);
    hungarian_final_reduce_kernel<<<1, THREADS, 0, stream>>>(row_sums, out, num_rows, inv_total);
}